// SurfnnconvNet_3393024164117
// MI455X (gfx1250) — compile-verified
//
#include <hip/hip_runtime.h>
#include <hip/hip_bf16.h>
#include <stdint.h>

typedef __attribute__((ext_vector_type(2))) float v2f;
typedef __attribute__((ext_vector_type(8))) float v8f;

#define N_NODES 50000
#define N_EDGES 100000

// ---------------------------------------------------------------------------
// zero-fill workspace (aggregation buffers must start at 0 every call)
// ---------------------------------------------------------------------------
__global__ void zero_f4(float4* __restrict__ p, int n4) {
  int i = blockIdx.x * blockDim.x + threadIdx.x;
  if (i < n4) p[i] = make_float4(0.f, 0.f, 0.f, 0.f);
}

// ---------------------------------------------------------------------------
// Fused NNConv edge pass:
//   W = relu(EA @ Wm + bm)            (WMMA f32 16x16x4, K=16 -> 4 steps)
//   msg[e,o] = sum_i x[src[e],i] * W[e, i*OUT_DIM + o]   (fused FMA)
//   aggr[dst[e], o] += msg[e,o]       (global_atomic_add_f32)
// One wave handles a 16-edge tile. Wm/bm staged in LDS, reused by all waves.
// C/D WMMA layout: lane l, VGPR v -> row M = v + 8*(l>>4), col N = l&15.
// A layout: lane l, VGPR v -> A[l&15][2*(l>>4) + v].
// ---------------------------------------------------------------------------
template<int IN_DIM, int OUT_DIM, int WPB>
__global__ __launch_bounds__(WPB * 32)
void edge_nnconv(const float* __restrict__ xfeat,      // [N, IN_DIM]
                 const int64_t* __restrict__ srcI,     // [E]
                 const int64_t* __restrict__ dstI,     // [E]
                 const float* __restrict__ ea,         // [E, 16]
                 const float* __restrict__ Wm,         // [16, IN_DIM*OUT_DIM]
                 const float* __restrict__ bm,         // [IN_DIM*OUT_DIM]
                 float* __restrict__ aggr,             // [N, OUT_DIM]
                 int nTiles)
{
  constexpr int NCOLS = IN_DIM * OUT_DIM;   // 2048 for both layers
  constexpr int NT    = NCOLS / 16;         // 128 column tiles
  constexpr int OCH   = OUT_DIM / 16;       // output chunks per input feature

  __shared__ float sWm[16 * NCOLS];         // 128 KB
  __shared__ float sBm[NCOLS];              // 8 KB
  __shared__ float sX[WPB][16 * IN_DIM];    // per-wave x[src] tiles
  __shared__ int   sDst[WPB][16];

  const int tid = threadIdx.x;
  for (int i = tid; i < 16 * NCOLS / 4; i += WPB * 32)
    ((float4*)sWm)[i] = ((const float4*)Wm)[i];
  for (int i = tid; i < NCOLS / 4; i += WPB * 32)
    ((float4*)sBm)[i] = ((const float4*)bm)[i];
  __syncthreads();

  const int wid  = tid >> 5;
  const int lane = tid & 31;
  const int tile = blockIdx.x * WPB + wid;
  if (tile >= nTiles) return;                 // wave-uniform: EXEC all-1 below
  const int e0 = tile * 16;
  const int m  = lane & 15;                   // row within tile / col N
  const int hi = lane >> 4;

  // stage x[src] tile: lane loads row m, half hi (IN_DIM/2 contiguous floats)
  {
    const int64_t s = srcI[e0 + m];
    const float* xr = xfeat + (size_t)s * IN_DIM + hi * (IN_DIM / 2);
    float* dr = &sX[wid][m * IN_DIM + hi * (IN_DIM / 2)];
#pragma unroll
    for (int j = 0; j < IN_DIM / 8; ++j)
      ((float4*)dr)[j] = ((const float4*)xr)[j];
    if (lane < 16) sDst[wid][lane] = (int)dstI[e0 + lane];
  }

  // A tile: edge_attr 16x16, split into 4 WMMA K-steps of 4
  v2f a[4];
#pragma unroll
  for (int kk = 0; kk < 4; ++kk)
    a[kk] = *(const v2f*)(ea + (size_t)(e0 + m) * 16 + 4 * kk + 2 * hi);

  v8f acc[OCH];
#pragma unroll
  for (int o = 0; o < OCH; ++o)
#pragma unroll
    for (int j = 0; j < 8; ++j) acc[o][j] = 0.f;

  float xsv[8];
#pragma unroll 4
  for (int t = 0; t < NT; ++t) {
    const int och = t % OCH;                  // resolved per unrolled body
    if (och == 0) {
      const int i = t / OCH;                  // input-feature index for chunk
#pragma unroll
      for (int v = 0; v < 8; ++v)
        xsv[v] = sX[wid][(v + 8 * hi) * IN_DIM + i];
    }
    const int c0 = t * 16;                    // c = i*OUT_DIM + o, contiguous
    const float bias = sBm[c0 + m];
    v8f c;
#pragma unroll
    for (int j = 0; j < 8; ++j) c[j] = bias;
#pragma unroll
    for (int kk = 0; kk < 4; ++kk) {
      v2f b;
      b.x = sWm[(4 * kk + 2 * hi)     * NCOLS + c0 + m];
      b.y = sWm[(4 * kk + 2 * hi + 1) * NCOLS + c0 + m];
      c = __builtin_amdgcn_wmma_f32_16x16x4_f32(false, a[kk], false, b,
                                                (short)0, c, false, false);
    }
#pragma unroll
    for (int j = 0; j < 8; ++j) c[j] = fmaxf(c[j], 0.f);   // edge-MLP ReLU
#pragma unroll
    for (int v = 0; v < 8; ++v)                            // fused contraction
      acc[och][v] = fmaf(xsv[v], c[v], acc[och][v]);
  }

  // scatter-add messages: aggr[dst[row], och*16 + n] += acc
#pragma unroll
  for (int v = 0; v < 8; ++v) {
    const int d = sDst[wid][v + 8 * hi];
    float* base = aggr + (size_t)d * OUT_DIM + m;
#pragma unroll
    for (int o = 0; o < OCH; ++o)
      unsafeAtomicAdd(base + o * 16, acc[o][v]);
  }
}

// ---------------------------------------------------------------------------
// h = relu(aggr1 + x @ Wr1 + b1)   (16-node tiles, K=32 -> 8 WMMA steps)
// ---------------------------------------------------------------------------
__global__ __launch_bounds__(256)
void node_root1(const float* __restrict__ xfeat,   // [N,32]
                const float* __restrict__ aggr1,   // [N,64]
                const float* __restrict__ Wr1,     // [32,64]
                const float* __restrict__ b1,      // [64]
                float* __restrict__ h,             // [N,64]
                int nTiles)
{
  __shared__ float sW[32 * 64];
  __shared__ float sB[64];
  const int tid = threadIdx.x;
  for (int i = tid; i < 32 * 64 / 4; i += 256)
    ((float4*)sW)[i] = ((const float4*)Wr1)[i];
  if (tid < 64) sB[tid] = b1[tid];
  __syncthreads();

  const int wid = tid >> 5, lane = tid & 31;
  const int tile = blockIdx.x * 8 + wid;
  if (tile >= nTiles) return;
  const int n0 = tile * 16;
  const int mrow = lane & 15, hi = lane >> 4;

  v2f a[8];
#pragma unroll
  for (int kk = 0; kk < 8; ++kk)
    a[kk] = *(const v2f*)(xfeat + (size_t)(n0 + mrow) * 32 + 4 * kk + 2 * hi);

#pragma unroll
  for (int tt = 0; tt < 4; ++tt) {
    const int c0 = tt * 16;
    const float bias = sB[c0 + mrow];
    v8f c;
#pragma unroll
    for (int v = 0; v < 8; ++v)
      c[v] = aggr1[(size_t)(n0 + v + 8 * hi) * 64 + c0 + mrow] + bias;
#pragma unroll
    for (int kk = 0; kk < 8; ++kk) {
      v2f b;
      b.x = sW[(4 * kk + 2 * hi)     * 64 + c0 + mrow];
      b.y = sW[(4 * kk + 2 * hi + 1) * 64 + c0 + mrow];
      c = __builtin_amdgcn_wmma_f32_16x16x4_f32(false, a[kk], false, b,
                                                (short)0, c, false, false);
    }
#pragma unroll
    for (int v = 0; v < 8; ++v)
      h[(size_t)(n0 + v + 8 * hi) * 64 + c0 + mrow] = fmaxf(c[v], 0.f);
  }
}

// ---------------------------------------------------------------------------
// h2 = aggr2 + h @ Wr2 + b2 ; out = h2 @ Wreg + breg
// ---------------------------------------------------------------------------
__global__ __launch_bounds__(256)
void node_root2_reg(const float* __restrict__ h,      // [N,64]
                    const float* __restrict__ aggr2,  // [N,32]
                    const float* __restrict__ Wr2,    // [64,32]
                    const float* __restrict__ b2,     // [32]
                    const float* __restrict__ Wreg,   // [32]
                    const float* __restrict__ breg,   // [1]
                    float* __restrict__ out,          // [N]
                    int nTiles)
{
  __shared__ float sW[64 * 32];
  __shared__ float sB[32];
  __shared__ float sWr[32];
  __shared__ float sH2[8][16][32];
  const int tid = threadIdx.x;
  for (int i = tid; i < 64 * 32 / 4; i += 256)
    ((float4*)sW)[i] = ((const float4*)Wr2)[i];
  if (tid < 32) { sB[tid] = b2[tid]; sWr[tid] = Wreg[tid]; }
  __syncthreads();

  const int wid = tid >> 5, lane = tid & 31;
  const int tile = blockIdx.x * 8 + wid;
  if (tile >= nTiles) return;
  const int n0 = tile * 16;
  const int mrow = lane & 15, hi = lane >> 4;

  v2f a[16];
#pragma unroll
  for (int kk = 0; kk < 16; ++kk)
    a[kk] = *(const v2f*)(h + (size_t)(n0 + mrow) * 64 + 4 * kk + 2 * hi);

#pragma unroll
  for (int tt = 0; tt < 2; ++tt) {
    const int c0 = tt * 16;
    const float bias = sB[c0 + mrow];
    v8f c;
#pragma unroll
    for (int v = 0; v < 8; ++v)
      c[v] = aggr2[(size_t)(n0 + v + 8 * hi) * 32 + c0 + mrow] + bias;
#pragma unroll
    for (int kk = 0; kk < 16; ++kk) {
      v2f b;
      b.x = sW[(4 * kk + 2 * hi)     * 32 + c0 + mrow];
      b.y = sW[(4 * kk + 2 * hi + 1) * 32 + c0 + mrow];
      c = __builtin_amdgcn_wmma_f32_16x16x4_f32(false, a[kk], false, b,
                                                (short)0, c, false, false);
    }
#pragma unroll
    for (int v = 0; v < 8; ++v)
      sH2[wid][v + 8 * hi][c0 + mrow] = c[v];
  }
  // same-wave LDS ops are issued in order; keep compiler from reordering
  __builtin_amdgcn_wave_barrier();
  if (lane < 16) {
    float accv = breg[0];
#pragma unroll
    for (int o = 0; o < 32; ++o)
      accv = fmaf(sH2[wid][lane][o], sWr[o], accv);
    out[n0 + lane] = accv;
  }
}

// ---------------------------------------------------------------------------
extern "C" void kernel_launch(void* const* d_in, const int* in_sizes, int n_in,
                              void* d_out, int out_size, void* d_ws, size_t ws_size,
                              hipStream_t stream)
{
  (void)in_sizes; (void)n_in; (void)out_size; (void)ws_size;
  const float*   x    = (const float*)d_in[0];
  const int64_t* ei   = (const int64_t*)d_in[1];
  const float*   ea   = (const float*)d_in[2];
  const float*   Wm1  = (const float*)d_in[3];
  const float*   bm1  = (const float*)d_in[4];
  const float*   Wm2  = (const float*)d_in[5];
  const float*   bm2  = (const float*)d_in[6];
  const float*   Wr1  = (const float*)d_in[7];
  const float*   b1   = (const float*)d_in[8];
  const float*   Wr2  = (const float*)d_in[9];
  const float*   b2   = (const float*)d_in[10];
  const float*   Wreg = (const float*)d_in[11];
  const float*   breg = (const float*)d_in[12];
  float* out = (float*)d_out;

  // workspace layout: aggr1 [N*64] | aggr2 [N*32] | h [N*64]  (32 MB total)
  float* aggr1 = (float*)d_ws;
  float* aggr2 = aggr1 + (size_t)N_NODES * 64;
  float* hbuf  = aggr2 + (size_t)N_NODES * 32;

  const int64_t* src = ei;
  const int64_t* dst = ei + N_EDGES;

  {
    int n4 = N_NODES * 96 / 4;   // zero aggr1 + aggr2 (contiguous)
    zero_f4<<<(n4 + 255) / 256, 256, 0, stream>>>((float4*)aggr1, n4);
  }

  const int eTiles = N_EDGES / 16;   // 6250 (exact)
  const int nTiles = N_NODES / 16;   // 3125 (exact)

  edge_nnconv<32, 64, 4><<<(eTiles + 3) / 4, 128, 0, stream>>>(
      x, src, dst, ea, Wm1, bm1, aggr1, eTiles);
  node_root1<<<(nTiles + 7) / 8, 256, 0, stream>>>(
      x, aggr1, Wr1, b1, hbuf, nTiles);
  edge_nnconv<64, 32, 4><<<(eTiles + 3) / 4, 128, 0, stream>>>(
      hbuf, src, dst, ea, Wm2, bm2, aggr2, eTiles);
  node_root2_reg<<<(nTiles + 7) / 8, 256, 0, stream>>>(
      hbuf, aggr2, Wr2, b2, Wreg, breg, out, nTiles);
}